// F1_67379446940315
// MI455X (gfx1250) — compile-verified
//
#include <hip/hip_runtime.h>
#include <hip/hip_bf16.h>

// ---------------------------------------------------------------------------
// F1-from-confusion-matrix, MI455X (gfx1250, wave32).
//
// Roofline: dominated by the one-shot 512 MB read of y_pred -> ~22us at
// 23.3 TB/s. Everything else (argmax VALU, 1M atomics, 128-class F1) hides
// under that stream. Main kernel: wave-per-row coalesced b128 NT loads +
// shfl_xor argmax reduction + one global atomic per row. Final kernel uses
// V_WMMA_F32_16X16X4_F32 (x2, chained) to sum the 128 f1 values.
// ---------------------------------------------------------------------------

typedef __attribute__((ext_vector_type(4))) float  vf4;
typedef __attribute__((ext_vector_type(2))) float  v2f;
typedef __attribute__((ext_vector_type(8))) float  v8f;

#define CLS 128
#define EPSF 1e-12f

// ---- kernel 1: zero the confusion matrix (must happen every launch) -------
__global__ void conf_zero_kernel(int* __restrict__ conf) {
    conf[blockIdx.x * blockDim.x + threadIdx.x] = 0;
}

// ---- kernel 2: streaming argmax + confusion scatter -----------------------
// One wave per row: lane l loads columns [4l, 4l+3] as a 16B b128 load, so a
// wave issues one contiguous 512B row. Non-temporal hint: 512MB stream,
// zero reuse, keep it out of L2's way.
__global__ __launch_bounds__(256) void argmax_conf_kernel(
    const float* __restrict__ y_pred,
    const int*   __restrict__ y_true,
    int*         __restrict__ conf,
    int n)
{
    const int lane   = threadIdx.x & 31;
    const int wave   = (int)((blockIdx.x * blockDim.x + threadIdx.x) >> 5);
    const int nwaves = (int)((gridDim.x * blockDim.x) >> 5);

    for (int row = wave; row < n; row += nwaves) {
        const vf4* p = reinterpret_cast<const vf4*>(y_pred + (size_t)row * CLS) + lane;
        vf4 v = __builtin_nontemporal_load(p);

        // lane-local argmax over 4 columns (strict > keeps first occurrence)
        int   base = lane * 4;
        float best = v.x; int bi = base;
        if (v.y > best) { best = v.y; bi = base + 1; }
        if (v.z > best) { best = v.z; bi = base + 2; }
        if (v.w > best) { best = v.w; bi = base + 3; }

        // wave32 xor-shuffle reduction; ties resolve to the smaller index
        #pragma unroll
        for (int off = 16; off > 0; off >>= 1) {
            float ov = __shfl_xor(best, off);
            int   oi = __shfl_xor(bi,   off);
            if (ov > best || (ov == best && oi < bi)) { best = ov; bi = oi; }
        }

        if (lane == 0) {
            int t = __builtin_nontemporal_load(y_true + row);
            atomicAdd(conf + t * CLS + bi, 1);
        }
    }
}

// ---- kernel 3: per-class F1 + WMMA sum reduction --------------------------
__device__ __forceinline__ float f1_of_class(const int* conf, int c) {
    float TP = (float)conf[c * CLS + c];
    // Faithful to the reference's quirky gather-index formulation:
    float FP = 127.0f * (float)conf[c * CLS + 1] + (float)conf[c * CLS + 0];
    float FN = 127.0f * (float)conf[1 * CLS + c] + (float)conf[0 * CLS + c];
    float sens = TP / (TP + FN + EPSF);
    float prec = TP / (TP + FP + EPSF);
    return 2.0f * (prec * sens / (prec + sens + EPSF));
}

__global__ void f1_finalize_kernel(const int* __restrict__ conf,
                                   float* __restrict__ out)
{
    const int lane = threadIdx.x;            // single wave, 0..31, EXEC all-1s
    // 32-bit A-matrix 16x4 layout: lanes 0-15 hold (M=lane, K=0..1),
    // lanes 16-31 hold (M=lane-16, K=2..3). Pack A[m][k] = f1[4m + k].
    const int m     = lane & 15;
    const int kbase = (lane >> 4) * 2;
    const int c0    = 4 * m + kbase;

    v2f a0, a1, b;
    a0.x = f1_of_class(conf, c0);            // classes   0..63  (WMMA #1)
    a0.y = f1_of_class(conf, c0 + 1);
    a1.x = f1_of_class(conf, c0 + 64);       // classes  64..127 (WMMA #2)
    a1.y = f1_of_class(conf, c0 + 65);
    b.x = 1.0f; b.y = 1.0f;                  // B = ones(4x16): D[m][n] = rowsum(A)[m]

    v8f acc = {};
    acc = __builtin_amdgcn_wmma_f32_16x16x4_f32(
        false, a0, false, b, (short)0, acc, false, false);
    acc = __builtin_amdgcn_wmma_f32_16x16x4_f32(
        false, a1, false, b, (short)0, acc, false, false);

    // D layout: lane 0 holds D[0..7][0] (classes 0..31 & 64..95),
    //           lane 16 holds D[8..15][0] (classes 32..63 & 96..127).
    float s = acc[0] + acc[1] + acc[2] + acc[3] + acc[4] + acc[5] + acc[6] + acc[7];
    s += __shfl_xor(s, 16);
    if (lane == 0) out[0] = s * (1.0f / (float)CLS);
}

// ---------------------------------------------------------------------------
extern "C" void kernel_launch(void* const* d_in, const int* in_sizes, int n_in,
                              void* d_out, int out_size, void* d_ws, size_t ws_size,
                              hipStream_t stream) {
    const float* y_pred = (const float*)d_in[0];
    const int*   y_true = (const int*)d_in[1];
    const int    n      = in_sizes[1];          // number of rows (N)
    int*   conf = (int*)d_ws;                    // 128*128 int32 = 64 KB scratch
    float* out  = (float*)d_out;

    conf_zero_kernel<<<(CLS * CLS) / 256, 256, 0, stream>>>(conf);
    argmax_conf_kernel<<<2048, 256, 0, stream>>>(y_pred, y_true, conf, n);
    f1_finalize_kernel<<<1, 32, 0, stream>>>(conf, out);
}